// MCPBRNN_Generic_PETconstraint_Three_VariantOutputGate_10196252360853
// MI455X (gfx1250) — compile-verified
//
#include <hip/hip_runtime.h>
#include <math.h>

// ---------------------------------------------------------------------------
// MCPBRNN scalar-recurrence kernel for gfx1250 (MI455X).
// Phase 1: latency-bound sequential scan (1 lane) -> carry trajectory,
//          written directly into the c_n output region.
// Phase 2: massively parallel gate/output recompute (B threads, ~36MB of
//          stores -> microseconds at 23.3 TB/s; the scan dominates).
// std(y_obs) reduction: per-wave 32->16 stage runs on the matrix pipe via
// v_wmma_f32_16x16x4_f32 (layout-agnostic all-ones-A column-sum), then a
// 4-level xor-shuffle tree; per-block partials folded in fixed order for
// bit-deterministic replays (no float atomics).
// ---------------------------------------------------------------------------

typedef float v2f __attribute__((ext_vector_type(2)));
typedef float v8f __attribute__((ext_vector_type(8)));

#define ML_C 2.9086f
#define SL_C 1.898f
#define LOG2E_C 1.4426950408889634f
#define SPIN_LEN 5000
#define TRAIN_LEN 400000
#define RED_BLOCKS 256
#define RED_THREADS 256

struct Params {
  const float* p_mean;
  const float* p_std;
  const float* w_yom;
  const float* w_yom_fp;
  const float* w_yom_gw;
  const float* w_ylm;
  const float* w_yfm;
  const float* w_b1_yom;
  const float* w_b1_yom_gw;
  const float* w_b1_yom_fp;
  const float* w_b2_ylm;
  const float* b0_yom;
  const float* b0_yom_gw;
  const float* b0_yom_fp;
  const float* b0_ylm;
  const int*   time_lag;
};

// Fully-folded per-gate coefficients: sigmoid(b0 + (c-mo)/so * b1) becomes
// rcp(1 + exp2(fma(c, A, C)))  with  A = -log2e*b1/so,  C = -log2e*b0 - mo*A.
struct Derived {
  float oo1, oogw1, oofp1, ol1;   // softmax fractions of the 5 rate params
  float Ao, Co, Agw, Cgw, Afp, Cfp;  // carry-gate sigmoid-arg coefficients
  float Al, Cl;                      // loss-gate (depends only on u2)
};

__device__ __forceinline__ Derived make_derived(const Params P) {
  Derived D;
  float eo  = __expf(P.w_yom[0]);
  float efp = __expf(P.w_yom_fp[0]);
  float egw = __expf(P.w_yom_gw[0]);
  float el  = __expf(P.w_ylm[0]);
  float ef  = __expf(P.w_yfm[0]);
  float inv_d = 1.0f / (eo + egw + el + ef + efp);
  D.oo1 = eo * inv_d; D.oogw1 = egw * inv_d; D.oofp1 = efp * inv_d; D.ol1 = el * inv_d;

  float mo = P.p_mean[0];
  float inv_so = 1.0f / P.p_std[0];

  D.Ao  = -LOG2E_C * P.w_b1_yom[0]    * inv_so;
  D.Co  = fmaf(-mo, D.Ao,  -LOG2E_C * P.b0_yom[0]);
  D.Agw = -LOG2E_C * P.w_b1_yom_gw[0] * inv_so;
  D.Cgw = fmaf(-mo, D.Agw, -LOG2E_C * P.b0_yom_gw[0]);
  D.Afp = -LOG2E_C * P.w_b1_yom_fp[0] * inv_so;
  D.Cfp = fmaf(-mo, D.Afp, -LOG2E_C * P.b0_yom_fp[0]);

  float k = P.w_b2_ylm[0] * (1.0f / SL_C);       // (u2-ML)/SL * b2l + b0l
  D.Al = -LOG2E_C * k;
  D.Cl = fmaf(-ML_C, D.Al, -LOG2E_C * P.b0_ylm[0]);
  return D;
}

// All gates for one step, given carry c0 and forcing u2.
// Per-gate dependent chain: fma -> v_exp_f32 -> add -> v_rcp_f32 -> mul.
__device__ __forceinline__ void step_gates(const Derived& D, float c0, float u2,
                                           float& oo, float& oogw, float& oofp,
                                           float& ol, float& olc, float& f) {
  float rc  = __builtin_amdgcn_rcpf(c0);                     // starts immediately
  float eo  = __builtin_amdgcn_exp2f(fmaf(c0, D.Ao,  D.Co));
  float egw = __builtin_amdgcn_exp2f(fmaf(c0, D.Agw, D.Cgw));
  float efp = __builtin_amdgcn_exp2f(fmaf(c0, D.Afp, D.Cfp));
  float el  = __builtin_amdgcn_exp2f(fmaf(u2, D.Al,  D.Cl)); // carry-independent
  oo   = D.oo1   * __builtin_amdgcn_rcpf(1.0f + eo);
  oogw = D.oogw1 * __builtin_amdgcn_rcpf(1.0f + egw);
  oofp = D.oofp1 * __builtin_amdgcn_rcpf(1.0f + efp);
  ol   = D.ol1   * __builtin_amdgcn_rcpf(1.0f + el);
  // ol - relu(ol - u2/c) == min(ol, u2/c)  (for c0 > 0)
  olc = (c0 > 0.0f) ? fminf(ol, u2 * rc) : ol;
  f = 1.0f - ((oo + oofp) + (oogw + olc));                   // balanced tree
}

// ---------------------------------------------------------------------------
// K1: grid-stride sum / sum-of-squares partials for std(y_obs[lo:hi]).
// Wave-level 32->16 reduction on the matrix pipe: A = all-ones 16x4, so
// D[m][n] = sum over K of column n of B. With B = {v, 0} per lane, column n
// holds {v[n], 0, v[n+16], 0} in *some* K order -> the all-ones dot yields
// v[n] + v[n+16] regardless of the hardware's B K-row arrangement, and D is
// replicated across rows so every lane holds a valid column sum.
// ---------------------------------------------------------------------------
__global__ void reduce_sum_sq_kernel(const float* __restrict__ y, int lo, int hi,
                                     float* __restrict__ partials) {
  int tid = blockIdx.x * blockDim.x + threadIdx.x;
  int stride = gridDim.x * blockDim.x;
  float s = 0.0f, q = 0.0f;
  for (int i = lo + tid; i < hi; i += stride) {
    float v = y[i];
    s += v;
    q = fmaf(v, v, q);
  }

  v2f a;  a[0] = 1.0f; a[1] = 1.0f;   // A = all ones (every row = (1,1,1,1))
  v2f bs; bs[0] = s;   bs[1] = 0.0f;
  v2f bq; bq[0] = q;   bq[1] = 0.0f;
  v8f cz = {};
  v8f ds_ = __builtin_amdgcn_wmma_f32_16x16x4_f32(false, a, false, bs,
                                                  (short)0, cz, false, false);
  v8f dq_ = __builtin_amdgcn_wmma_f32_16x16x4_f32(false, a, false, bq,
                                                  (short)0, cz, false, false);
  float sh = ds_[0];   // lane L holds s[L&15] + s[(L&15)+16]
  float qh = dq_[0];
#pragma unroll
  for (int m = 8; m >= 1; m >>= 1) {
    sh += __shfl_xor(sh, m, 32);
    qh += __shfl_xor(qh, m, 32);
  }

  __shared__ float sS[RED_THREADS / 32];
  __shared__ float sQ[RED_THREADS / 32];
  unsigned lane = threadIdx.x & 31u;
  int wv = threadIdx.x >> 5;
  if (lane == 0) { sS[wv] = sh; sQ[wv] = qh; }
  __syncthreads();
  if (threadIdx.x == 0) {
    float bsum = 0.0f, bqsum = 0.0f;
    int nw = blockDim.x >> 5;
    for (int w = 0; w < nw; ++w) { bsum += sS[w]; bqsum += sQ[w]; }
    partials[2 * blockIdx.x]     = bsum;   // fixed slot: deterministic
    partials[2 * blockIdx.x + 1] = bqsum;
  }
}

// K2: fold partials (fixed order) -> ws[0] = std with ddof=1.
__global__ void finalize_std_kernel(const float* __restrict__ partials, int nblocks,
                                    float n_count, float* __restrict__ scal) {
  if (blockIdx.x == 0 && threadIdx.x == 0) {
    float s = 0.0f, q = 0.0f;
    for (int i = 0; i < nblocks; ++i) {
      s += partials[2 * i];
      q += partials[2 * i + 1];
    }
    float obsstd = 0.0f;
    if (n_count > 1.5f) {
      float mean = s / n_count;
      float var  = (q - s * mean) / (n_count - 1.0f);
      obsstd = sqrtf(fmaxf(var, 0.0f));
    }
    scal[0] = obsstd;
  }
}

// ---------------------------------------------------------------------------
// K3: the sequential scan (the runtime floor: ~B dependent iterations).
// Chunked by 16 steps: one cacheline prefetch per chunk at +8KB, all 16
// float2 loads issued before the dependent chain, body unrolled.
// ---------------------------------------------------------------------------
__global__ void scan_kernel(const float* __restrict__ x, Params P,
                            float* __restrict__ c_out, int B) {
  if (blockIdx.x != 0 || threadIdx.x != 0) return;
  Derived D = make_derived(P);
  int tl = P.time_lag[0];
  if (tl < 0) tl = 0;
  if (tl > B) tl = B;

  const float2* __restrict__ xp = (const float2*)x;
  float c0 = 0.0f;
  int i = tl;
  for (; i + 16 <= B; i += 16) {
    if (i + 1024 < B) {
      __builtin_prefetch(&xp[i + 1024], 0, 3);   // global_prefetch_b8, 8KB ahead
    }
    float2 u[16];
#pragma unroll
    for (int k = 0; k < 16; ++k) u[k] = xp[i + k];
#pragma unroll
    for (int k = 0; k < 16; ++k) {
      c_out[i + k] = c0;                         // c_n[t] = carry BEFORE step t
      float oo, oogw, oofp, ol, olc, f;
      step_gates(D, c0, u[k].y, oo, oogw, oofp, ol, olc, f);
      c0 = fmaf(f, c0, u[k].x);                  // c1 = f*c0 + u1
    }
  }
  for (; i < B; ++i) {
    float2 u = xp[i];
    c_out[i] = c0;
    float oo, oogw, oofp, ol, olc, f;
    step_gates(D, c0, u.y, oo, oogw, oofp, ol, olc, f);
    c0 = fmaf(f, c0, u.x);
  }
}

// ---------------------------------------------------------------------------
// K4: parallel output materialization. Each thread t recomputes the gates
// from the stored carry and writes all 17 output columns.
// Output layout (flat, in tuple return order):
//   [0:B)h  [B:2B)hfp  [2B:3B)c  [3B:4B)l  [4B:5B)lc  [5B:6B)bp=0
//   [6B:7B)gw  [7B:8B)Gib=0  [8B:9B)Goo  [9B:10B)Goofp  [10B:11B)Gol
//   [11B:12B)Golc  [12B:13B)Gf  [13B:14B)Googw
//   [14B:16B) h_nout (B,2 interleaved)   [16B:17B) obs_std
// ---------------------------------------------------------------------------
__global__ void outputs_kernel(const float* __restrict__ x,
                               const float* __restrict__ scal, Params P,
                               float* __restrict__ out, int B) {
  int t = blockIdx.x * blockDim.x + threadIdx.x;
  if (t >= B) return;

  size_t Bz = (size_t)B;
  float* r_h    = out;
  float* r_hfp  = out + Bz;
  float* r_c    = out + 2 * Bz;   // already holds the carry trajectory
  float* r_l    = out + 3 * Bz;
  float* r_lc   = out + 4 * Bz;
  float* r_bp   = out + 5 * Bz;
  float* r_gw   = out + 6 * Bz;
  float* r_gib  = out + 7 * Bz;
  float* r_goo  = out + 8 * Bz;
  float* r_gfp  = out + 9 * Bz;
  float* r_gol  = out + 10 * Bz;
  float* r_golc = out + 11 * Bz;
  float* r_gf   = out + 12 * Bz;
  float* r_gogw = out + 13 * Bz;
  float* r_hno  = out + 14 * Bz;  // (B,2) interleaved
  float* r_ostd = out + 16 * Bz;

  int tl = P.time_lag[0];
  if (tl < 0) tl = 0;
  if (tl > B) tl = B;

  if (t < tl) {
    r_h[t] = 0.0f;  r_hfp[t] = 0.0f;  r_c[t] = 0.0f;   r_l[t] = 0.0f;
    r_lc[t] = 0.0f; r_bp[t] = 0.0f;   r_gw[t] = 0.0f;  r_gib[t] = 0.0f;
    r_goo[t] = 0.0f; r_gfp[t] = 0.0f; r_gol[t] = 0.0f; r_golc[t] = 0.0f;
    r_gf[t] = 0.0f;  r_gogw[t] = 0.0f;
    r_hno[2 * (size_t)t] = 0.0f; r_hno[2 * (size_t)t + 1] = 0.0f;
    r_ostd[t] = 0.0f;
    return;
  }

  Derived D = make_derived(P);
  float c0 = r_c[t];               // carry written by scan_kernel
  float u2 = x[2 * (size_t)t + 1];

  float oo, oogw, oofp, ol, olc, f;
  step_gates(D, c0, u2, oo, oogw, oofp, ol, olc, f);

  float obsstd = scal[0];
  float h = oo * c0;

  r_h[t]    = h;
  r_hfp[t]  = oofp * c0;
  /* r_c[t] already correct */
  r_l[t]    = ol * c0;
  r_lc[t]   = olc * c0;
  r_bp[t]   = 0.0f;
  r_gw[t]   = oogw * c0;
  r_gib[t]  = 0.0f;
  r_goo[t]  = oo;
  r_gfp[t]  = oofp;
  r_gol[t]  = ol;
  r_golc[t] = olc;
  r_gf[t]   = f;
  r_gogw[t] = oogw;
  r_hno[2 * (size_t)t]     = h;
  r_hno[2 * (size_t)t + 1] = obsstd;
  r_ostd[t] = obsstd;
}

// ---------------------------------------------------------------------------
extern "C" void kernel_launch(void* const* d_in, const int* in_sizes, int n_in,
                              void* d_out, int out_size, void* d_ws, size_t ws_size,
                              hipStream_t stream) {
  const float* x     = (const float*)d_in[0];
  const float* y_obs = (const float*)d_in[1];

  Params P;
  P.p_mean      = (const float*)d_in[2];
  P.p_std       = (const float*)d_in[3];
  // d_in[4] = epoch (unused by reference)
  P.time_lag    = (const int*)d_in[5];
  P.w_yom       = (const float*)d_in[6];
  P.w_yom_fp    = (const float*)d_in[7];
  P.w_yom_gw    = (const float*)d_in[8];
  P.w_ylm       = (const float*)d_in[9];
  P.w_yfm       = (const float*)d_in[10];
  P.w_b1_yom    = (const float*)d_in[11];
  P.w_b1_yom_gw = (const float*)d_in[12];
  P.w_b1_yom_fp = (const float*)d_in[13];
  P.w_b2_ylm    = (const float*)d_in[14];
  P.b0_yom      = (const float*)d_in[15];
  P.b0_yom_gw   = (const float*)d_in[16];
  P.b0_yom_fp   = (const float*)d_in[17];
  P.b0_ylm      = (const float*)d_in[18];

  int B = in_sizes[1];                 // y_obs is (B,1)
  float* out = (float*)d_out;
  float* ws  = (float*)d_ws;           // ws[0]=obsstd, ws[16..16+2*RED_BLOCKS)=partials

  int lo = SPIN_LEN  < B ? SPIN_LEN  : B;
  int hi = TRAIN_LEN < B ? TRAIN_LEN : B;

  reduce_sum_sq_kernel<<<RED_BLOCKS, RED_THREADS, 0, stream>>>(y_obs, lo, hi, ws + 16);
  finalize_std_kernel<<<1, 32, 0, stream>>>(ws + 16, RED_BLOCKS, (float)(hi - lo), ws);
  scan_kernel<<<1, 32, 0, stream>>>(x, P, out + 2 * (size_t)B, B);
  outputs_kernel<<<(B + 255) / 256, 256, 0, stream>>>(x, ws, P, out, B);
}